// GCN_3496103379117
// MI455X (gfx1250) — compile-verified
//
#include <hip/hip_runtime.h>
#include <hip/hip_bf16.h>

// Problem constants (from reference)
#define BB   16
#define FF   64
#define NN   512
#define MM   16
#define CCAT 320
#define COUT 64

typedef float v2f __attribute__((ext_vector_type(2)));
typedef float v8f __attribute__((ext_vector_type(8)));

// D = A(16x4 f32) x B(4x16 f32) + C(16x16 f32)
static __device__ __forceinline__ v8f wmma4(v2f a, v2f b, v8f c) {
    return __builtin_amdgcn_wmma_f32_16x16x4_f32(
        /*neg_a=*/false, a, /*neg_b=*/false, b,
        /*c_mod=*/(short)0, c, /*reuse_a=*/false, /*reuse_b=*/false);
}

// ---- operand loaders -------------------------------------------------------
// A-operand (16x4 f32) from an LDS matrix S[64][512] at (row f0.., cols k0..k0+3)
// ISA layout: lanes 0-15 hold M=0..15; VGPR0 = {K=0 | K=2}, VGPR1 = {K=1 | K=3}
static __device__ __forceinline__ v2f ldsA_tile(const float* S, int f0, int k0, int lane) {
    int row  = f0 + (lane & 15);
    int kb   = k0 + ((lane >> 4) ? 2 : 0);
    const float* p = S + row * NN + kb;
    v2f r; r.x = p[0]; r.y = p[1];
    return r;
}

// B-operand (4x16 f32) from global adjacency G[512][512] at (rows k0.., cols v0..)
static __device__ __forceinline__ v2f glbB_tile(const float* __restrict__ G, int k0, int v0, int lane) {
    int col = v0 + (lane & 15);
    int kb  = k0 + ((lane >> 4) ? 2 : 0);
    v2f r;
    r.x = G[(size_t)kb * NN + col];
    r.y = G[(size_t)(kb + 1) * NN + col];
    return r;
}

// B-operand (4x16 f32) from an LDS slot matrix S[64][512], K dimension = f
static __device__ __forceinline__ v2f ldsB_tile(const float* S, int k0, int v0, int lane) {
    int col = v0 + (lane & 15);
    int kb  = k0 + ((lane >> 4) ? 2 : 0);
    v2f r;
    r.x = S[kb * NN + col];
    r.y = S[(kb + 1) * NN + col];
    return r;
}

// A-operand (16x4 f32) from global W[64][320] at (rows o0.., cols c0..c0+3)
static __device__ __forceinline__ v2f glbW_tile(const float* __restrict__ Wg, int o0, int c0, int lane) {
    int row = o0 + (lane & 15);
    int cb  = c0 + ((lane >> 4) ? 2 : 0);
    const float* p = Wg + row * CCAT + cb;
    v2f r; r.x = p[0]; r.y = p[1];
    return r;
}

// Store a 16x16 f32 C/D tile into LDS T[64][512] at (f0.., v0..)
// ISA layout: VGPR r: lanes 0-15 -> M=r, lanes 16-31 -> M=r+8
static __device__ __forceinline__ void store_tile_lds(float* T, int f0, int v0, int lane, v8f acc) {
    int rowadd = (lane >> 4) ? 8 : 0;
    int col    = v0 + (lane & 15);
#pragma unroll
    for (int r = 0; r < 8; ++r)
        T[(f0 + r + rowadd) * NN + col] = acc[r];
}

// ---- diffusion GEMM: dst[64, v0..v0+64) = src(64x512, LDS) @ Ag(512x512, global)
static __device__ __forceinline__ void diffuse(const float* src, float* dst,
                                               const float* __restrict__ Ag,
                                               int v0, int lane) {
#pragma unroll
    for (int vtg = 0; vtg < 2; ++vtg) {
        int vbase = v0 + vtg * 32;
        v8f acc[4][2];
#pragma unroll
        for (int ft = 0; ft < 4; ++ft)
#pragma unroll
            for (int vt = 0; vt < 2; ++vt)
#pragma unroll
                for (int r = 0; r < 8; ++r) acc[ft][vt][r] = 0.0f;

        for (int k0 = 0; k0 < NN; k0 += 4) {
            if (k0 + 8 < NN)  // pull next K-panel of the adjacency toward L2/L0
                __builtin_prefetch(Ag + (size_t)(k0 + 8) * NN + vbase, 0, 0);
            v2f a[4], bm[2];
#pragma unroll
            for (int ft = 0; ft < 4; ++ft) a[ft] = ldsA_tile(src, ft * 16, k0, lane);
#pragma unroll
            for (int vt = 0; vt < 2; ++vt) bm[vt] = glbB_tile(Ag, k0, vbase + vt * 16, lane);
#pragma unroll
            for (int ft = 0; ft < 4; ++ft)
#pragma unroll
                for (int vt = 0; vt < 2; ++vt)
                    acc[ft][vt] = wmma4(a[ft], bm[vt], acc[ft][vt]);
        }
#pragma unroll
        for (int ft = 0; ft < 4; ++ft)
#pragma unroll
            for (int vt = 0; vt < 2; ++vt)
                store_tile_lds(dst, ft * 16, vbase + vt * 16, lane, acc[ft][vt]);
    }
}

// ---- Y accumulation: Y[64, v0..v0+64) += W[:, 64*slot .. +64] @ S(64x512, LDS)
static __device__ __forceinline__ void ymix(v8f (&Y)[4][4], const float* __restrict__ Wg,
                                            int slot, const float* S, int v0, int lane) {
    int cbase = slot * 64;
#pragma unroll 4
    for (int k0 = 0; k0 < 64; k0 += 4) {
        v2f a[4], bm[4];
#pragma unroll
        for (int ot = 0; ot < 4; ++ot) a[ot] = glbW_tile(Wg, ot * 16, cbase + k0, lane);
#pragma unroll
        for (int vt = 0; vt < 4; ++vt) bm[vt] = ldsB_tile(S, k0, v0 + vt * 16, lane);
#pragma unroll
        for (int ot = 0; ot < 4; ++ot)
#pragma unroll
            for (int vt = 0; vt < 4; ++vt)
                Y[ot][vt] = wmma4(a[ot], bm[vt], Y[ot][vt]);
    }
}

__global__ __launch_bounds__(256, 1)
void gcn_fused(const float* __restrict__ x,  const float* __restrict__ a0,
               const float* __restrict__ a1, const float* __restrict__ Wg,
               const float* __restrict__ bias, float* __restrict__ y) {
    extern __shared__ float smem[];
    float* B0 = smem;               // 64*512 floats = 128 KB
    float* B1 = smem + FF * NN;     // 128 KB

    const int bm   = blockIdx.x;        // b*16 + m
    const int b    = bm >> 4;
    const int m    = bm & 15;
    const int lane = threadIdx.x & 31;
    const int wave = threadIdx.x >> 5;
    const int v0   = wave * 64;

    const float* Ag0 = a0 + (size_t)bm * NN * NN;   // a0[b][m]
    const float* Ag1 = a1 + (size_t)bm * NN * NN;
    const float* xg  = x + (size_t)b * FF * NN * MM + m;   // x[b][f][n][m]

    // cooperative strided load of X slice -> B0
    for (int i = threadIdx.x; i < FF * NN; i += 256) {
        int f = i >> 9, n = i & (NN - 1);
        B0[i] = xg[((size_t)f * NN + n) * MM];
    }
    __syncthreads();

    // Y accumulator, initialized with bias broadcast along v
    v8f Y[4][4];
    {
        int rowadd = (lane >> 4) ? 8 : 0;
#pragma unroll
        for (int ot = 0; ot < 4; ++ot) {
#pragma unroll
            for (int r = 0; r < 8; ++r) {
                float bv = bias[ot * 16 + r + rowadd];
#pragma unroll
                for (int vt = 0; vt < 4; ++vt) Y[ot][vt][r] = bv;
            }
        }
    }

    ymix(Y, Wg, 0, B0, v0, lane);           // slot 0: X itself
    diffuse(B0, B1, Ag0, v0, lane);         // X1a = X @ A0
    __syncthreads();
    ymix(Y, Wg, 1, B1, v0, lane);
    diffuse(B1, B0, Ag0, v0, lane);         // X2a = X1a @ A0 (X sacrificed)
    __syncthreads();
    ymix(Y, Wg, 2, B0, v0, lane);

    // reload X -> B1 (B1 fully consumed before the last barrier)
    for (int i = threadIdx.x; i < FF * NN; i += 256) {
        int f = i >> 9, n = i & (NN - 1);
        B1[i] = xg[((size_t)f * NN + n) * MM];
    }
    __syncthreads();

    diffuse(B1, B0, Ag1, v0, lane);         // X1b = X @ A1
    __syncthreads();
    ymix(Y, Wg, 3, B0, v0, lane);
    diffuse(B0, B1, Ag1, v0, lane);         // X2b = X1b @ A1
    __syncthreads();
    ymix(Y, Wg, 4, B1, v0, lane);

    // ReLU + scatter store into y[b][o][n][m]
    float* yg = y + (size_t)b * COUT * NN * MM + m;
    {
        int rowadd = (lane >> 4) ? 8 : 0;
        int cl     = lane & 15;
#pragma unroll
        for (int ot = 0; ot < 4; ++ot)
#pragma unroll
            for (int vt = 0; vt < 4; ++vt)
#pragma unroll
                for (int r = 0; r < 8; ++r) {
                    int o = ot * 16 + r + rowadd;
                    int n = v0 + vt * 16 + cl;
                    float val = Y[ot][vt][r];
                    yg[((size_t)o * NN + n) * MM] = fmaxf(val, 0.0f);
                }
    }
}

extern "C" void kernel_launch(void* const* d_in, const int* in_sizes, int n_in,
                              void* d_out, int out_size, void* d_ws, size_t ws_size,
                              hipStream_t stream) {
    (void)in_sizes; (void)n_in; (void)out_size; (void)d_ws; (void)ws_size;
    const float* x  = (const float*)d_in[0];
    const float* a0 = (const float*)d_in[1];
    const float* a1 = (const float*)d_in[2];
    const float* W  = (const float*)d_in[3];
    const float* b  = (const float*)d_in[4];
    float* y = (float*)d_out;

    const size_t shmem = (size_t)2 * FF * NN * sizeof(float);  // 256 KB of the 320 KB WGP LDS
    dim3 grid(BB * MM);   // 256 independent (b,m) sub-problems
    dim3 block(256);      // 8 wave32s, one 64-column output stripe each
    gcn_fused<<<grid, block, shmem, stream>>>(x, a0, a1, W, b, y);
}